// StickyHDPHMMVI_20727512170621
// MI455X (gfx1250) — compile-verified
//
#include <hip/hip_runtime.h>
#include <hip/hip_bf16.h>

// ---------------- problem constants ----------------
constexpr int Dd   = 64;    // feature dim
constexpr int KP1  = 33;    // mixture components
constexpr int Rr   = 8;     // low-rank columns
constexpr int NPAD = 48;    // K padded to 3 WMMA n-tiles
constexpr int QPAIR = (Dd * (Dd + 1)) / 2;   // 2080 upper-tri Gram entries
constexpr int QMU0  = QPAIR;                 // 64 mu entries
constexpr int QONE  = QPAIR + Dd;            // 2144: constant row
constexpr int QTOT  = ((QONE + 1 + 7) / 8) * 8;  // 2152, multiple of 8 (dual-chain)

constexpr int BT_TILE = 16;   // (b,t) rows per block == WMMA M
constexpr int CHUNK   = 512;  // inner-dim chunk staged in LDS (mult of 8)
constexpr int YSTR    = Dd * (Rr + 1) + 1;   // 577 (pad to break bank conflicts)
constexpr int VSTR    = Dd + 1;              // 65

typedef float v2f __attribute__((ext_vector_type(2)));
typedef float v8f __attribute__((ext_vector_type(8)));

// ---------------- digamma (args here are in [~1.5, ~36]) ----------------
__device__ __forceinline__ float digammaf_dev(float x) {
    float r = 0.0f;
    while (x < 6.0f) { r -= 1.0f / x; x += 1.0f; }
    float f = 1.0f / (x * x);
    return r + logf(x) - 0.5f / x
           - f * (1.0f / 12.0f - f * (1.0f / 120.0f - f * (1.0f / 252.0f)));
}

// =======================================================================
// Kernel 1: per-k emission cache -> packed GEMM B-matrix in workspace.
//   blocks 0..47  : column k of Bmat (k>=33 -> zeros)
//   block  48     : pair-index decode table
// Bmat layout: element (q,k) at ((q>>1)*NPAD + k)*2 + (q&1)  -> the per-lane
// WMMA B fragment (two consecutive K values for one column) is one b64 load.
// =======================================================================
__global__ __launch_bounds__(64) void emis_prep_kernel(
    const float* __restrict__ mu_k, const float* __restrict__ Psi,
    const float* __restrict__ nu,   const float* __restrict__ kappa,
    float* __restrict__ Bws, unsigned int* __restrict__ table)
{
    const int k   = blockIdx.x;
    const int tid = threadIdx.x;

    if (k == NPAD) {  // build (d,e) decode table
        for (int q = tid; q < QTOT; q += 64) {
            unsigned int enc;
            if (q < QPAIR) {
                int d = 0, p = q;
                while (p >= (Dd - d)) { p -= (Dd - d); ++d; }
                int e = d + p;
                enc = (unsigned)d | ((unsigned)e << 8);              // type 0
            } else if (q < QONE) {
                enc = (unsigned)(q - QMU0) | (1u << 16);             // mu row
            } else if (q == QONE) {
                enc = (2u << 16);                                    // const row
            } else {
                enc = (3u << 16);                                    // zero pad
            }
            table[q] = enc;
        }
        return;
    }
    if (k >= KP1) {   // zero-pad columns 33..47 (ws is poisoned!)
        for (int q = tid; q < QTOT; q += 64)
            Bws[((q >> 1) * NPAD + k) * 2 + (q & 1)] = 0.0f;
        return;
    }

    constexpr int LD = Dd + 1;
    __shared__ float P [Dd * LD];   // Psi -> L -> E_Lam
    __shared__ float Wm[Dd * LD];   // L^{-1}
    __shared__ float vsh[Dd];
    __shared__ float red[Dd];
    __shared__ float dsh[Dd];
    __shared__ float base_sh;

    for (int i = tid; i < Dd * Dd; i += 64) {
        int r = i >> 6, c = i & 63;
        P[r * LD + c] = Psi[k * Dd * Dd + i];
    }
    __syncthreads();

    // Cholesky, in-place lower triangle
    for (int j = 0; j < Dd; ++j) {
        if (tid == 0) {
            float s = P[j * LD + j];
            for (int t = 0; t < j; ++t) { float l = P[j * LD + t]; s -= l * l; }
            P[j * LD + j] = sqrtf(s);
        }
        __syncthreads();
        int i = tid;
        if (i > j) {
            float s = P[i * LD + j];
            for (int t = 0; t < j; ++t) s -= P[i * LD + t] * P[j * LD + t];
            P[i * LD + j] = s / P[j * LD + j];
        }
        __syncthreads();
    }
    red[tid] = logf(P[tid * LD + tid]);   // for logdet(Psi) = 2*sum

    // W = L^{-1}: each thread owns one column (forward substitution)
    {
        const int c = tid;
        Wm[c * LD + c] = 1.0f / P[c * LD + c];
        for (int i = c + 1; i < Dd; ++i) {
            float s = 0.0f;
            for (int t = c; t < i; ++t) s += P[i * LD + t] * Wm[t * LD + c];
            Wm[i * LD + c] = -s / P[i * LD + i];
        }
    }
    __syncthreads();

    const float nuk = nu[k];
    // E_Lam = nu * W^T W  (overwrite P; L no longer needed)
    {
        const int d = tid;
        for (int e = 0; e < Dd; ++e) {
            int i0 = (d > e) ? d : e;
            float s = 0.0f;
            for (int i = i0; i < Dd; ++i) s += Wm[i * LD + d] * Wm[i * LD + e];
            P[d * LD + e] = nuk * s;
        }
    }
    __syncthreads();

    // v = E_Lam @ mu_k ; digamma terms
    {
        const int d = tid;
        float s = 0.0f;
        for (int e = 0; e < Dd; ++e) s += P[d * LD + e] * mu_k[k * Dd + e];
        vsh[d] = s;
    }
    dsh[tid] = digammaf_dev((nuk + 1.0f - (float)(tid + 1)) * 0.5f);
    __syncthreads();

    if (tid == 0) {
        float logdet = 0.0f, digsum = 0.0f, c2 = 0.0f;
        for (int i = 0; i < Dd; ++i) {
            logdet += red[i]; digsum += dsh[i];
            c2 += vsh[i] * mu_k[k * Dd + i];
        }
        logdet *= 2.0f;
        const float LOG2   = 0.69314718055994531f;
        const float LOG2PI = 1.83787706640934534f;
        float E_logdet = digsum + (float)Dd * LOG2 - logdet;
        float cst  = 0.5f * (E_logdet - (float)Dd * LOG2PI);
        base_sh = cst - 0.5f * (c2 + (float)Dd / kappa[k]);
    }
    __syncthreads();

    // emit packed column k:  [w_de * S[d,e]] (2080) | [-2 v] (64) | [-2 base] | 0
    for (int q = tid; q < QTOT; q += 64) {
        float val;
        if (q < QPAIR) {
            int d = 0, p = q;
            while (p >= (Dd - d)) { p -= (Dd - d); ++d; }
            int e = d + p;
            float s = P[d * LD + e];
            val = (d == e) ? s : 2.0f * s;    // symmetry weight
        } else if (q < QONE) {
            val = -2.0f * vsh[q - QMU0];
        } else if (q == QONE) {
            val = -2.0f * base_sh;
        } else {
            val = 0.0f;
        }
        Bws[((q >> 1) * NPAD + k) * 2 + (q & 1)] = val;
    }
}

// =======================================================================
// Kernel 2: per 16 (b,t) rows -- build Gram-based A rows in LDS chunks,
// run the 16x48 x 2152 GEMM with V_WMMA_F32_16X16X4_F32.
// Two independent accumulator chains per wave (even/odd inner segments)
// to keep >=2 WMMAs in flight per wave.   logB[bt,k] = -0.5*acc (masked)
// =======================================================================
__global__ __launch_bounds__(96) void emis_main_kernel(
    const float* __restrict__ mu_t, const float* __restrict__ diag_var,
    const float* __restrict__ F_t,  const unsigned char* __restrict__ mask,
    const float* __restrict__ Bws,  const unsigned int* __restrict__ table,
    float* __restrict__ out)
{
    __shared__ float Ylds[BT_TILE * YSTR];               // [mu | F] per bt row
    __shared__ float Vlds[BT_TILE * VSTR];               // diag variances
    __shared__ float msk[BT_TILE];
    __shared__ __align__(16) float Alds[CHUNK * BT_TILE]; // A fragments, chunked

    const int tid = threadIdx.x;
    const int bt0 = blockIdx.x * BT_TILE;

    // ---- phase 0: stage Y = [mu | F], var, mask ----
    for (int i = tid; i < BT_TILE * Dd; i += 96) {
        int m = i >> 6, d = i & 63;
        Ylds[m * YSTR + d * 9] = mu_t[(bt0 + m) * Dd + d];
        Vlds[m * VSTR + d]     = diag_var[(bt0 + m) * Dd + d];
    }
    const float4* F4 = (const float4*)F_t;               // F rows: 8 floats
    for (int i = tid; i < BT_TILE * Dd * 2; i += 96) {   // 2048 float4s
        float4 f  = F4[(size_t)bt0 * (Dd * Rr / 4) + i];
        int base  = i * 4;
        int m = base >> 9, j = base & 511;
        int d = j >> 3,    r = j & 7;
        float* p = &Ylds[m * YSTR + d * 9 + 1 + r];
        p[0] = f.x; p[1] = f.y; p[2] = f.z; p[3] = f.w;
    }
    if (tid < BT_TILE) msk[tid] = mask[bt0 + tid] ? 1.0f : 0.0f;  // jax bool = 1B
    __syncthreads();

    const int w    = tid >> 5;     // wave id: owns k-tile [16w, 16w+16)
    const int lane = tid & 31;
    const int ml   = lane & 15;    // A row (M) / B column (N) for this lane
    const int kh   = lane >> 4;    // K-half select per WMMA fragment layout
    v8f acc0 = {0.f, 0.f, 0.f, 0.f, 0.f, 0.f, 0.f, 0.f};
    v8f acc1 = {0.f, 0.f, 0.f, 0.f, 0.f, 0.f, 0.f, 0.f};

    for (int qbase = 0; qbase < QTOT; qbase += CHUNK) {
        int csize = QTOT - qbase;
        if (csize > CHUNK) csize = CHUNK;                 // last chunk = 104 (mult of 8)

        // ---- build A chunk: A[m][q] in WMMA fragment layout ----
        for (int idx = tid; idx < BT_TILE * csize; idx += 96) {
            int m  = idx & 15;
            int qq = idx >> 4;
            unsigned enc = table[qbase + qq];
            int typ = enc >> 16;
            float val;
            if (typ == 0) {                               // Gram upper-tri entry
                int d = enc & 255, e = (enc >> 8) & 255;
                const float* ya = &Ylds[m * YSTR + d * 9];
                const float* yb = &Ylds[m * YSTR + e * 9];
                float s = (d == e) ? Vlds[m * VSTR + d] : 0.0f;  // fold tr_diag
                #pragma unroll
                for (int c = 0; c < 9; ++c) s += ya[c] * yb[c];
                val = s;
            } else if (typ == 1) {                        // mu row
                val = Ylds[m * YSTR + (int)(enc & 255) * 9];
            } else if (typ == 2) {                        // constant row
                val = 1.0f;
            } else {
                val = 0.0f;
            }
            Alds[(qq >> 1) * 32 + (m << 1) + (qq & 1)] = val;
        }
        __syncthreads();

        // ---- consume: two independent WMMA chains per wave ----
        for (int q0 = 0; q0 < csize; q0 += 8) {
            int qg = qbase + q0;
            v2f a0 = *(const v2f*)&Alds[((q0 >> 1) + kh) * 32 + (ml << 1)];
            v2f b0 = *(const v2f*)&Bws[(((qg >> 1) + kh) * NPAD + (w * 16 + ml)) * 2];
            acc0 = __builtin_amdgcn_wmma_f32_16x16x4_f32(
                false, a0, false, b0, (short)0, acc0, false, false);
            v2f a1 = *(const v2f*)&Alds[(((q0 + 4) >> 1) + kh) * 32 + (ml << 1)];
            v2f b1 = *(const v2f*)&Bws[((((qg + 4) >> 1) + kh) * NPAD + (w * 16 + ml)) * 2];
            acc1 = __builtin_amdgcn_wmma_f32_16x16x4_f32(
                false, a1, false, b1, (short)0, acc1, false, false);
        }
        __syncthreads();
    }

    // ---- store: C/D layout -> lane holds col n=ml, rows m=j+8*kh ----
    const int kcol = w * 16 + ml;
    if (kcol < KP1) {
        v8f acc = acc0 + acc1;
        #pragma unroll
        for (int j = 0; j < 8; ++j) {
            int m = j + 8 * kh;
            out[(size_t)(bt0 + m) * KP1 + kcol] = -0.5f * acc[j] * msk[m];
        }
    }
}

// =======================================================================
extern "C" void kernel_launch(void* const* d_in, const int* in_sizes, int n_in,
                              void* d_out, int out_size, void* d_ws, size_t ws_size,
                              hipStream_t stream) {
    const float* mu_t  = (const float*)d_in[0];
    const float* dvar  = (const float*)d_in[1];
    const float* F_t   = (const float*)d_in[2];
    const float* mu_k  = (const float*)d_in[3];
    const float* Psi   = (const float*)d_in[4];
    const float* nu    = (const float*)d_in[5];
    const float* kappa = (const float*)d_in[6];
    const unsigned char* mask = (const unsigned char*)d_in[7]; // jax bool: 1B
    float* out = (float*)d_out;

    // workspace: Bmat (QTOT*NPAD floats = 413,184 B) | pair table (QTOT u32)
    float* Bws = (float*)d_ws;
    unsigned int* table =
        (unsigned int*)((char*)d_ws + (size_t)QTOT * NPAD * sizeof(float));

    const int BT = in_sizes[0] / Dd;   // B*T = 32768

    emis_prep_kernel<<<NPAD + 1, 64, 0, stream>>>(mu_k, Psi, nu, kappa, Bws, table);
    emis_main_kernel<<<BT / BT_TILE, 96, 0, stream>>>(
        mu_t, dvar, F_t, mask, Bws, table, out);
}